// Cat_sp_tensor_90950227460820
// MI455X (gfx1250) — compile-verified
//
#include <hip/hip_runtime.h>

// Problem constants fixed by the reference (pseudo_image_dims = [1024,1024], B=4)
#define HW_SHIFT 20                        // H*W = 1<<20
#define W_SHIFT  10                        // W   = 1<<10
#define KEY_SPACE (4u << 20)               // B*H*W = 4,194,304 distinct keys
#define NWORDS (KEY_SPACE / 32u)           // 131,072 bitmap words (512 KB)
#define SCAN_BLOCK 256
#define NBLOCKS_SCAN (NWORDS / SCAN_BLOCK) // 512

typedef __attribute__((ext_vector_type(4))) float f32x4;

#if __has_builtin(__builtin_amdgcn_global_load_async_to_lds_b32) && \
    __has_builtin(__builtin_amdgcn_s_wait_asynccnt)
#define HAVE_ASYNC_LDS 1
typedef __attribute__((address_space(1))) int g_int;  // HIP prints AS1 as __device__
typedef __attribute__((address_space(3))) int l_int;  // LDS
#define AS1I(p) ((g_int*)(p))
#define AS3I(p) ((l_int*)(p))
#endif

// ---------------------------------------------------------------------------
// Shared-memory Hillis-Steele scan; returns the EXCLUSIVE prefix of `val`.
// blockDim.x must be a power of two and equal the shared array length.
// ---------------------------------------------------------------------------
__device__ __forceinline__ unsigned block_excl_scan(unsigned val, unsigned* sm) {
  const int t = threadIdx.x;
  const int n = blockDim.x;
  sm[t] = val;
  __syncthreads();
  for (int off = 1; off < n; off <<= 1) {
    unsigned x = (t >= off) ? sm[t - off] : 0u;
    __syncthreads();
    sm[t] += x;
    __syncthreads();
  }
  return sm[t] - val; // inclusive -> exclusive
}

// ---------------------------------------------------------------------------
// Step 1: one presence bitmap per tensor (atomicOr idempotent -> determin.)
// ---------------------------------------------------------------------------
__global__ void mark_keys_kernel(const int* __restrict__ idx1, int n1,
                                 const int* __restrict__ idx2, int n2,
                                 unsigned* __restrict__ bm1,
                                 unsigned* __restrict__ bm2) {
  const int i = blockIdx.x * blockDim.x + threadIdx.x;
  const int* idx;
  unsigned* bm;
  int r;
  if (i < n1)            { idx = idx1; bm = bm1; r = i; }
  else if (i < n1 + n2)  { idx = idx2; bm = bm2; r = i - n1; }
  else return;
  const unsigned b = (unsigned)idx[r * 3 + 0];
  const unsigned y = (unsigned)idx[r * 3 + 1];
  const unsigned x = (unsigned)idx[r * 3 + 2];
  const unsigned key = (b << HW_SHIFT) | (y << W_SHIFT) | x;
  atomicOr(&bm[key >> 5], 1u << (key & 31u));
}

// ---------------------------------------------------------------------------
// Step 2a: per-word exclusive scan of popcounts of the UNION bitmap within
// 256-word blocks. Bitmap words are staged into LDS with the gfx1250 async
// global->LDS path (ASYNCcnt) when available, else plain loads.
// ---------------------------------------------------------------------------
__global__ void scan_words_kernel(const unsigned* __restrict__ bm1,
                                  const unsigned* __restrict__ bm2,
                                  unsigned* __restrict__ wordPrefix,
                                  unsigned* __restrict__ blockSums) {
  __shared__ unsigned sm[SCAN_BLOCK];
  const int t = threadIdx.x;
  const int w = blockIdx.x * SCAN_BLOCK + t;
  unsigned uni;
#if HAVE_ASYNC_LDS
  __shared__ unsigned w1[SCAN_BLOCK];
  __shared__ unsigned w2[SCAN_BLOCK];
  __builtin_amdgcn_global_load_async_to_lds_b32(AS1I(bm1 + w), AS3I(&w1[t]), 0, 0);
  __builtin_amdgcn_global_load_async_to_lds_b32(AS1I(bm2 + w), AS3I(&w2[t]), 0, 0);
  __builtin_amdgcn_s_wait_asynccnt(0);
  __syncthreads();
  uni = w1[t] | w2[t];
#else
  uni = bm1[w] | bm2[w];
#endif
  const unsigned cnt  = (unsigned)__popc(uni);
  const unsigned excl = block_excl_scan(cnt, sm);
  wordPrefix[w] = excl;
  if (t == SCAN_BLOCK - 1) blockSums[blockIdx.x] = excl + cnt;
}

// ---------------------------------------------------------------------------
// Step 2b: exclusive scan of the 512 block totals (in place) + total U.
// ---------------------------------------------------------------------------
__global__ void scan_block_sums_kernel(unsigned* __restrict__ blockSums,
                                       unsigned* __restrict__ totalU) {
  __shared__ unsigned sm[NBLOCKS_SCAN];
  const unsigned v    = blockSums[threadIdx.x];
  const unsigned excl = block_excl_scan(v, sm);
  blockSums[threadIdx.x] = excl; // each thread touches only its own slot
  if (threadIdx.x == NBLOCKS_SCAN - 1) *totalU = excl + v;
}

// ---------------------------------------------------------------------------
// Step 3: scatter one feature row (64 f32 = 256 B) with 16 lanes x float4.
// rank(key) = #distinct union-keys < key = sorted-unique inverse index.
// If the key is absent from the OTHER tensor, this kernel also zero-fills the
// complementary half, so every output byte in [0,U) rows is written once.
// ---------------------------------------------------------------------------
__global__ void scatter_kernel(const float* __restrict__ feat,
                               const int* __restrict__ idx, int n,
                               const unsigned* __restrict__ bmSelf,
                               const unsigned* __restrict__ bmOther,
                               const unsigned* __restrict__ wordPrefix,
                               const unsigned* __restrict__ blockOffs,
                               float* __restrict__ out,
                               int selfOff, int othOff) {
  const long long gt = (long long)blockIdx.x * blockDim.x + threadIdx.x;
  const int row = (int)(gt >> 4);
  const int c   = (int)(gt & 15);
  if (row >= n) return;

  const unsigned b = (unsigned)idx[row * 3 + 0];
  const unsigned y = (unsigned)idx[row * 3 + 1];
  const unsigned x = (unsigned)idx[row * 3 + 2];
  const unsigned key = (b << HW_SHIFT) | (y << W_SHIFT) | x;
  const unsigned w   = key >> 5;
  const unsigned bit = key & 31u;

  const unsigned wSelf = bmSelf[w];
  const unsigned wOth  = bmOther[w];
  const unsigned uni   = wSelf | wOth;
  const unsigned rank  = wordPrefix[w] + blockOffs[w >> 8] +
                         (unsigned)__popc(uni & ((1u << bit) - 1u));

  const f32x4 v = __builtin_nontemporal_load(
      (const f32x4*)(feat + (size_t)row * 64) + c);
  float* rowBase = out + (size_t)rank * 128;
  __builtin_nontemporal_store(v, (f32x4*)(rowBase + selfOff) + c);
  if (!((wOth >> bit) & 1u)) {
    const f32x4 z = {0.f, 0.f, 0.f, 0.f};
    __builtin_nontemporal_store(z, (f32x4*)(rowBase + othOff) + c);
  }
}

// ---------------------------------------------------------------------------
// Step 4: zero only the tail rows [U, n_tot). 32 lanes x float4 per row.
// Threads below U exit after one uniform scalar load.
// ---------------------------------------------------------------------------
__global__ void zero_tail_kernel(const unsigned* __restrict__ totalU, int ntot,
                                 float* __restrict__ out) {
  const long long gt = (long long)blockIdx.x * blockDim.x + threadIdx.x;
  const int row = (int)(gt >> 5);
  const int c   = (int)(gt & 31);
  if (row >= ntot) return;
  const unsigned U = *totalU; // uniform -> s_load
  if ((unsigned)row < U) return;
  const f32x4 z = {0.f, 0.f, 0.f, 0.f};
  __builtin_nontemporal_store(z, (f32x4*)(out + (size_t)row * 128) + c);
}

// ---------------------------------------------------------------------------
// Host launcher (graph-capture safe: hipMemsetAsync + kernel launches only).
// d_ws layout: [bm1|bm2: 2*NWORDS u32][wordPrefix: NWORDS u32]
//              [blockSums: 512 u32][totalU: 1 u32]   (~1.57 MB)
// ---------------------------------------------------------------------------
extern "C" void kernel_launch(void* const* d_in, const int* in_sizes, int n_in,
                              void* d_out, int out_size, void* d_ws, size_t ws_size,
                              hipStream_t stream) {
  const float* f1 = (const float*)d_in[0];
  const float* f2 = (const float*)d_in[1];
  const int*   i1 = (const int*)d_in[2];
  const int*   i2 = (const int*)d_in[3];
  const int n1 = in_sizes[0] / 64;  // C1 = 64
  const int n2 = in_sizes[1] / 64;  // C2 = 64
  float* out = (float*)d_out;
  const int ntot = out_size / 128;  // output rows

  unsigned* bm1        = (unsigned*)d_ws;
  unsigned* bm2        = bm1 + NWORDS;
  unsigned* wordPrefix = bm2 + NWORDS;
  unsigned* blockSums  = wordPrefix + NWORDS;
  unsigned* totalU     = blockSums + NBLOCKS_SCAN;

  // Zero both bitmaps (no state may be carried between calls)
  (void)hipMemsetAsync(bm1, 0, 2u * NWORDS * sizeof(unsigned), stream);

  const int nm = n1 + n2;
  mark_keys_kernel<<<(nm + 255) / 256, 256, 0, stream>>>(i1, n1, i2, n2, bm1, bm2);

  scan_words_kernel<<<NBLOCKS_SCAN, SCAN_BLOCK, 0, stream>>>(bm1, bm2,
                                                             wordPrefix, blockSums);
  scan_block_sums_kernel<<<1, NBLOCKS_SCAN, 0, stream>>>(blockSums, totalU);

  {
    const long long t = (long long)n1 * 16;
    scatter_kernel<<<(unsigned)((t + 255) / 256), 256, 0, stream>>>(
        f1, i1, n1, bm1, bm2, wordPrefix, blockSums, out, 0, 64);
  }
  {
    const long long t = (long long)n2 * 16;
    scatter_kernel<<<(unsigned)((t + 255) / 256), 256, 0, stream>>>(
        f2, i2, n2, bm2, bm1, wordPrefix, blockSums, out, 64, 0);
  }
  {
    const long long t = (long long)ntot * 32;
    zero_tail_kernel<<<(unsigned)((t + 255) / 256), 256, 0, stream>>>(
        totalU, ntot, out);
  }
}